// GATNetHeadsChanged3LayersLeakyReLU_31628139168053
// MI455X (gfx1250) — compile-verified
//
#include <hip/hip_runtime.h>
#include <math.h>

#define NN 10000
#define EEDGE 160000
#define FIN 512
#define HHEAD 2
#define CCH 256
#define HC 512   // HHEAD*CCH

typedef float v2f __attribute__((ext_vector_type(2)));
typedef float v8f __attribute__((ext_vector_type(8)));

__device__ __forceinline__ float leakyf(float x, float s) { return x >= 0.f ? x : s * x; }

// ---------------------------------------------------------------------------
// fp32 WMMA GEMM: C[M,N] = act(A[M,K] @ B[K,N] + bias), row-major.
// One wave32 computes a 16x32 output tile = two 16x16 WMMA accumulators that
// share one A fragment -> two independent v_wmma per k-step (breaks the C->D
// serial chain) and half the A traffic. 8 waves / 256-thread block.
// M,K multiples of 16; N multiple of 32 (512/256/128 all qualify).
// ---------------------------------------------------------------------------
__global__ void gemm_wmma_f32(const float* __restrict__ A, const float* __restrict__ B,
                              const float* __restrict__ bias, float* __restrict__ C,
                              int M, int N, int K, float slope, int act) {
  int wave = threadIdx.x >> 5;
  int lane = threadIdx.x & 31;
  int tile = blockIdx.x * 8 + wave;          // one 16x32 tile per wave
  int nDT = N >> 5;                          // double-tiles per row of C
  int ntiles = (M >> 4) * nDT;
  if (tile >= ntiles) return;                // wave-uniform: EXEC all-ones for WMMA
  int tn = tile % nDT;
  int tm = tile / nDT;
  int l15 = lane & 15;
  int klo = (lane >> 4) << 1;                // lanes 16..31 hold K+2, K+3

  const float* arow = A + (size_t)(tm * 16 + l15) * K + klo;
  const float* bp0 = B + tn * 32 + l15;      // first 16-col sub-tile
  const float* bp1 = bp0 + 16;               // second 16-col sub-tile

  v8f acc0 = {}, acc1 = {};
  for (int k = 0; k < K; k += 4) {
    v2f a = *(const v2f*)(arow + k);         // A[M=l15][k+klo .. k+klo+1]
    size_t r0 = (size_t)(k + klo) * N;
    size_t r1 = r0 + N;
    v2f b0, b1;
    b0.x = bp0[r0]; b0.y = bp0[r1];
    b1.x = bp1[r0]; b1.y = bp1[r1];
    acc0 = __builtin_amdgcn_wmma_f32_16x16x4_f32(false, a, false, b0,
                                                 (short)0, acc0, false, false);
    acc1 = __builtin_amdgcn_wmma_f32_16x16x4_f32(false, a, false, b1,
                                                 (short)0, acc1, false, false);
  }

  int col0 = tn * 32 + l15;
  int col1 = col0 + 16;
  float bv0 = bias ? bias[col0] : 0.f;
  float bv1 = bias ? bias[col1] : 0.f;
  int row0 = tm * 16 + ((lane >> 4) << 3);   // lanes>=16 write rows M+8..M+15
#pragma unroll
  for (int r = 0; r < 8; ++r) {
    float v0 = acc0[r] + bv0;
    float v1 = acc1[r] + bv1;
    if (act) { v0 = leakyf(v0, slope); v1 = leakyf(v1, slope); }
    size_t base = (size_t)(row0 + r) * N;
    C[base + col0] = v0;
    C[base + col1] = v1;
  }
}

// ---------------------------------------------------------------------------
// Per (node, head): a_s = <h, att_src>, a_d = <h, att_dst>; seed segment max
// with the self-loop logit. One wave per (node,head) pair.
// ---------------------------------------------------------------------------
__global__ void node_prep(const float* __restrict__ h, const float* __restrict__ att_src,
                          const float* __restrict__ att_dst, float* __restrict__ a_s,
                          float* __restrict__ a_d, float* __restrict__ m) {
  int wave = threadIdx.x >> 5, lane = threadIdx.x & 31;
  int pair = blockIdx.x * 8 + wave;                  // pair = i*2 + head
  if (pair >= NN * HHEAD) return;
  int i = pair >> 1, hh = pair & 1;
  const float* hp = h + (size_t)i * HC + hh * CCH;
  const float* as = att_src + hh * CCH;
  const float* ad = att_dst + hh * CCH;
  float ps = 0.f, pd = 0.f;
  for (int c = lane; c < CCH; c += 32) {
    float hv = hp[c];
    ps += hv * as[c];
    pd += hv * ad[c];
  }
#pragma unroll
  for (int o = 16; o > 0; o >>= 1) {
    ps += __shfl_xor(ps, o, 32);
    pd += __shfl_xor(pd, o, 32);
  }
  if (lane == 0) {
    a_s[pair] = ps;
    a_d[pair] = pd;
    m[pair] = leakyf(ps + pd, 0.2f);                 // self-loop seeds the max
  }
}

// Segment max over edges via f32 atomic max (L2 atomics).
__global__ void edge_max(const int* __restrict__ src, const int* __restrict__ dst,
                         const float* __restrict__ a_s, const float* __restrict__ a_d,
                         float* __restrict__ m) {
  int e = blockIdx.x * blockDim.x + threadIdx.x;
  if (e >= EEDGE) return;
  int s = src[e], d = dst[e];
#pragma unroll
  for (int hh = 0; hh < 2; ++hh) {
    float ev = leakyf(a_s[s * 2 + hh] + a_d[d * 2 + hh], 0.2f);
    atomicMax(&m[d * 2 + hh], ev);
  }
}

// Self-loop exp term; initializes the softmax denominator.
__global__ void node_self(const float* __restrict__ a_s, const float* __restrict__ a_d,
                          const float* __restrict__ m, float* __restrict__ eeself,
                          float* __restrict__ denom) {
  int p = blockIdx.x * blockDim.x + threadIdx.x;
  if (p >= NN * HHEAD) return;
  float es = leakyf(a_s[p] + a_d[p], 0.2f);
  float v = expf(es - m[p]);
  eeself[p] = v;
  denom[p] = v;
}

// Per-edge exp + denominator accumulation; stores unnormalized ee per edge.
__global__ void edge_expsum(const int* __restrict__ src, const int* __restrict__ dst,
                            const float* __restrict__ a_s, const float* __restrict__ a_d,
                            const float* __restrict__ m, float* __restrict__ ee,
                            float* __restrict__ denom) {
  int e = blockIdx.x * blockDim.x + threadIdx.x;
  if (e >= EEDGE) return;
  int s = src[e], d = dst[e];
#pragma unroll
  for (int hh = 0; hh < 2; ++hh) {
    float ev = leakyf(a_s[s * 2 + hh] + a_d[d * 2 + hh], 0.2f);
    float x = expf(ev - m[d * 2 + hh]);
    ee[e * 2 + hh] = x;
    atomicAdd(&denom[d * 2 + hh], x);
  }
}

// agg = eeself * h  (self-loop message, unnormalized)
__global__ void agg_init(const float* __restrict__ h, const float* __restrict__ eeself,
                         float* __restrict__ agg) {
  size_t t = (size_t)blockIdx.x * 256 + threadIdx.x;
  if (t >= (size_t)NN * HC) return;
  int i = (int)(t / HC);
  int k = (int)(t % HC);
  int hh = k >> 8;
  agg[t] = eeself[i * 2 + hh] * h[t];
}

// agg[dst] += ee * h[src]  — one block per edge, threads over channels (coalesced).
__global__ void edge_agg(const int* __restrict__ src, const int* __restrict__ dst,
                         const float* __restrict__ h, const float* __restrict__ ee,
                         float* __restrict__ agg) {
  int e = blockIdx.x;
  int c = threadIdx.x;                               // 0..255
  int s = src[e], d = dst[e];
#pragma unroll
  for (int hh = 0; hh < 2; ++hh) {
    float w = ee[e * 2 + hh];
    atomicAdd(&agg[(size_t)d * HC + hh * CCH + c],
              w * h[(size_t)s * HC + hh * CCH + c]);
  }
}

// out = agg/denom + gat_b ; t0 = leaky(out, 0.01)
__global__ void post_kernel(const float* __restrict__ agg, const float* __restrict__ denom,
                            const float* __restrict__ gat_b, float* __restrict__ t0) {
  size_t t = (size_t)blockIdx.x * 256 + threadIdx.x;
  if (t >= (size_t)NN * HC) return;
  int i = (int)(t / HC);
  int k = (int)(t % HC);
  int hh = k >> 8;
  float v = agg[t] / (denom[i * 2 + hh] + 1e-16f) + gat_b[k];
  t0[t] = leakyf(v, 0.01f);
}

// y3 = y2 @ W2 + b2 (N=3), plus squared norm packed into .w
__global__ void tail_kernel(const float* __restrict__ y2, const float* __restrict__ W2,
                            const float* __restrict__ b2, float4* __restrict__ y3) {
  int i = blockIdx.x * blockDim.x + threadIdx.x;
  if (i >= NN) return;
  float a0 = b2[0], a1 = b2[1], a2 = b2[2];
  const float* r = y2 + (size_t)i * 128;
  for (int k = 0; k < 128; ++k) {
    float v = r[k];
    a0 += v * W2[k * 3 + 0];
    a1 += v * W2[k * 3 + 1];
    a2 += v * W2[k * 3 + 2];
  }
  y3[i] = make_float4(a0, a1, a2, a0 * a0 + a1 * a1 + a2 * a2);
}

// d[i,j] = sqrt(max(sq_i + sq_j - 2<y_i,y_j>, 0)) — reference formula.
// Each lane produces 4 adjacent columns -> one b128 store (store-bound kernel).
__global__ void cdist_kernel(const float4* __restrict__ y3, float* __restrict__ out) {
  int j4 = blockIdx.x * 256 + threadIdx.x;           // 2500 quads per row
  int i = blockIdx.y;
  if (j4 >= NN / 4) return;
  int j = j4 * 4;
  float4 yi = y3[i];
  float4 r;
  float* rp = &r.x;
#pragma unroll
  for (int q = 0; q < 4; ++q) {
    float4 yj = y3[j + q];
    float d2 = yi.w + yj.w - 2.f * (yi.x * yj.x + yi.y * yj.y + yi.z * yj.z);
    d2 = fmaxf(d2, 0.f);
    rp[q] = d2 > 0.f ? sqrtf(d2) : 0.f;
  }
  *(float4*)(out + (size_t)i * NN + j) = r;
}

extern "C" void kernel_launch(void* const* d_in, const int* in_sizes, int n_in,
                              void* d_out, int out_size, void* d_ws, size_t ws_size,
                              hipStream_t stream) {
  (void)in_sizes; (void)n_in; (void)out_size; (void)ws_size;
  const float* x       = (const float*)d_in[0];
  const int*   ei      = (const int*)d_in[1];
  const float* gat_W   = (const float*)d_in[2];
  const float* att_src = (const float*)d_in[3];
  const float* att_dst = (const float*)d_in[4];
  const float* gat_b   = (const float*)d_in[5];
  const float* Wa      = (const float*)d_in[6];
  const float* ba      = (const float*)d_in[7];
  const float* W1      = (const float*)d_in[8];
  const float* b1      = (const float*)d_in[9];
  const float* W2      = (const float*)d_in[10];
  const float* b2      = (const float*)d_in[11];
  float* out = (float*)d_out;

  const int* src = ei;
  const int* dst = ei + EEDGE;

  // Workspace carving (floats)
  float* ws = (float*)d_ws;
  float* h      = ws;                    ws += (size_t)NN * HC;       // 5.12M
  float* agg    = ws;                    ws += (size_t)NN * HC;       // 5.12M
  float* t0     = ws;                    ws += (size_t)NN * HC;       // 5.12M
  float* y1     = ws;                    ws += (size_t)NN * 256;      // 2.56M
  float* y2     = ws;                    ws += (size_t)NN * 128;      // 1.28M
  float* a_s    = ws;                    ws += NN * HHEAD;
  float* a_d    = ws;                    ws += NN * HHEAD;
  float* m      = ws;                    ws += NN * HHEAD;
  float* eeself = ws;                    ws += NN * HHEAD;
  float* denom  = ws;                    ws += NN * HHEAD;
  float* ee     = ws;                    ws += (size_t)EEDGE * HHEAD; // 320K
  float4* y3    = (float4*)ws;           ws += (size_t)NN * 4;

  // 1) h = x @ gat_W          (625 * 16 = 10000 16x32 tiles / 8 waves)
  gemm_wmma_f32<<<1250, 256, 0, stream>>>(x, gat_W, nullptr, h, NN, HC, FIN, 1.f, 0);
  // 2) attention logits + self-loop max seed (20000 wave-pairs / 8)
  node_prep<<<2500, 256, 0, stream>>>(h, att_src, att_dst, a_s, a_d, m);
  // 3) segment max over edges
  edge_max<<<(EEDGE + 255) / 256, 256, 0, stream>>>(src, dst, a_s, a_d, m);
  // 4) self-loop exp, denom init
  node_self<<<(NN * HHEAD + 255) / 256, 256, 0, stream>>>(a_s, a_d, m, eeself, denom);
  // 5) per-edge exp + denom accumulation
  edge_expsum<<<(EEDGE + 255) / 256, 256, 0, stream>>>(src, dst, a_s, a_d, m, ee, denom);
  // 6) unnormalized aggregation, self term
  agg_init<<<(NN * HC + 255) / 256, 256, 0, stream>>>(h, eeself, agg);
  // 7) edge scatter-add
  edge_agg<<<EEDGE, 256, 0, stream>>>(src, dst, h, ee, agg);
  // 8) normalize + bias + leaky(0.01)
  post_kernel<<<(NN * HC + 255) / 256, 256, 0, stream>>>(agg, denom, gat_b, t0);
  // 9) y1 = leaky(t0 @ Wa + ba)   (625 * 8 = 5000 tiles)
  gemm_wmma_f32<<<625, 256, 0, stream>>>(t0, Wa, ba, y1, NN, 256, 512, 0.01f, 1);
  // 10) y2 = leaky(y1 @ W1 + b1)  (625 * 4 = 2500 tiles, guarded)
  gemm_wmma_f32<<<313, 256, 0, stream>>>(y1, W1, b1, y2, NN, 128, 256, 0.01f, 1);
  // 11) y3 = y2 @ W2 + b2, packed with squared norms
  tail_kernel<<<(NN + 255) / 256, 256, 0, stream>>>(y2, W2, b2, y3);
  // 12) 10000x10000 distance matrix (400 MB store-bound), float4 stores
  cdist_kernel<<<dim3((NN / 4 + 255) / 256, NN), 256, 0, stream>>>(y3, out);
}